// Memcodes_16870631539145
// MI455X (gfx1250) — compile-verified
//
#include <hip/hip_runtime.h>
#include <hip/hip_bf16.h>
#include <cstdint>

// Problem constants (reference: B, N, H, D, C = 2, 4096, 16, 64, 1024)
#define BB 2
#define NN 4096
#define HH 16
#define DD 64
#define CC 1024

typedef __attribute__((ext_vector_type(2))) float v2f;
typedef __attribute__((ext_vector_type(8))) float v8f;
typedef __attribute__((ext_vector_type(4))) int   v4i;

// D = A(16x4 f32) * B(4x16 f32) + C(16x16 f32)
// 8-arg form: (neg_a, A, neg_b, B, c_mod, C, reuse_a, reuse_b)
__device__ __forceinline__ v8f wmma_f32x4(v2f a, v2f b, v8f c) {
    return __builtin_amdgcn_wmma_f32_16x16x4_f32(
        /*neg_a=*/false, a, /*neg_b=*/false, b,
        /*c_mod=*/(short)0, c, /*reuse_a=*/false, /*reuse_b=*/false);
}

// --- CDNA5 async global->LDS staging (guarded; falls back to ds staging) ----
#if defined(__gfx1250__) && __has_builtin(__builtin_amdgcn_global_load_async_to_lds_b128)
#define USE_ASYNC_LDS 1
// Prototype (from clang diagnostic): arg0 = v4i __device__* (addrspace 1),
// arg1 = v4i __shared__* (addrspace 3), then imm offset, imm cpol.
typedef __attribute__((address_space(1))) v4i ga_v4i;
typedef __attribute__((address_space(3))) v4i ls_v4i;
#if __has_builtin(__builtin_amdgcn_s_wait_asynccnt)
#define WAIT_ASYNC0() __builtin_amdgcn_s_wait_asynccnt(0)
#else
#define WAIT_ASYNC0() asm volatile("s_wait_asynccnt 0x0" ::: "memory")
#endif
__device__ __forceinline__ void async_copy16(const float* gsrc, float* ldst) {
    __builtin_amdgcn_global_load_async_to_lds_b128(
        (ga_v4i*)(uintptr_t)gsrc,
        (ls_v4i*)(uint32_t)(uintptr_t)ldst, 0, 0);
}
#else
#define USE_ASYNC_LDS 0
#endif

// ---------------------------------------------------------------------------
// Kernel 1: per-head projections  k = codebooks @ wk,  v = codebooks @ wv
// codebooks: (H, C, D); wk/wv: (H, D, D) -> kbuf/vbuf: (H, C, D)
// One wave handles a 16-row tile of one head; A-fragments shared for k and v.
// ---------------------------------------------------------------------------
__global__ __launch_bounds__(256) void memcodes_project(
    const float* __restrict__ cb, const float* __restrict__ wk,
    const float* __restrict__ wv, float* __restrict__ kbuf,
    float* __restrict__ vbuf)
{
    const int tid  = threadIdx.x;
    const int wave = tid >> 5;
    const int lane = tid & 31;
    const int n    = lane & 15;
    const int half = lane >> 4;

    const int tile    = blockIdx.x * 8 + wave;   // 0 .. H*64-1
    const int h       = tile >> 6;
    const int rowbase = (tile & 63) * 16;

    // A fragments: codebook rows (16 x 64), K-chunks of 4
    v2f afrag[16];
    const float* ap = cb + ((size_t)h * CC + rowbase + n) * DD;
#pragma unroll
    for (int kk = 0; kk < 16; ++kk)
        afrag[kk] = *(const v2f*)(ap + 4 * kk + 2 * half);

    const float* wkh = wk + (size_t)h * DD * DD;
    const float* wvh = wv + (size_t)h * DD * DD;

#pragma unroll
    for (int jt = 0; jt < 4; ++jt) {             // 4 column tiles of 16
        v8f ck = {};
        v8f cv = {};
#pragma unroll
        for (int kk = 0; kk < 16; ++kk) {
            const int dk = 4 * kk + 2 * half;
            v2f bk, bv;
            bk.x = wkh[(size_t)dk * DD + jt * 16 + n];
            bk.y = wkh[(size_t)(dk + 1) * DD + jt * 16 + n];
            bv.x = wvh[(size_t)dk * DD + jt * 16 + n];
            bv.y = wvh[(size_t)(dk + 1) * DD + jt * 16 + n];
            ck = wmma_f32x4(afrag[kk], bk, ck);
            cv = wmma_f32x4(afrag[kk], bv, cv);
        }
#pragma unroll
        for (int r = 0; r < 8; ++r) {
            const size_t o =
                ((size_t)h * CC + rowbase + r + 8 * half) * DD + jt * 16 + n;
            kbuf[o] = ck[r];
            vbuf[o] = cv[r];
        }
    }
}

// ---------------------------------------------------------------------------
// Kernel 2: logits = (q * d^-0.5) @ k^T ; streaming argmax over codes;
//           gather v rows; emit out, indices, per-head histogram.
// Block = 256 threads = 8 waves; each wave owns 16 q-rows of one (b,h).
// k-tiles are double-buffered through LDS via async global->LDS copies
// (ASYNCcnt-tracked) when available.
// ---------------------------------------------------------------------------
__global__ __launch_bounds__(256) void memcodes_main(
    const float* __restrict__ x, const float* __restrict__ kbuf,
    const float* __restrict__ vbuf, float* __restrict__ out,
    float* __restrict__ idx_out, unsigned int* __restrict__ hist)
{
    __shared__ __align__(16) float ktile[2][16 * DD];   // 2 x 4 KB tiles

    const int b    = blockIdx.z;
    const int h    = blockIdx.y;
    const int tid  = threadIdx.x;
    const int wave = tid >> 5;
    const int lane = tid & 31;
    const int n    = lane & 15;
    const int half = lane >> 4;
    const int rowbase = blockIdx.x * 128 + wave * 16;

    // Load + pre-scale q A-fragments (16 rows x 64, scale = D^-0.5 = 0.125)
    v2f afrag[16];
    {
        const float* qp =
            x + (((size_t)b * NN + rowbase + n) * (HH * DD) + (size_t)h * DD);
        const float scale = 0.125f;
#pragma unroll
        for (int kk = 0; kk < 16; ++kk) {
            v2f t = *(const v2f*)(qp + 4 * kk + 2 * half);
            afrag[kk] = t * scale;
        }
    }

    float best[8];
    int   bestj[8];
#pragma unroll
    for (int r = 0; r < 8; ++r) { best[r] = -3.402823e38f; bestj[r] = 0; }

    const float* kh = kbuf + (size_t)h * CC * DD;

#if USE_ASYNC_LDS
    // Each thread copies 16B of the tile: 256 threads x 16B = 4 KB.
    async_copy16(kh + (size_t)tid * 4, &ktile[0][tid * 4]);
#endif

    for (int jt = 0; jt < CC / 16; ++jt) {       // 64 code tiles
        const int cur = jt & 1;
#if USE_ASYNC_LDS
        WAIT_ASYNC0();          // this thread's copy of tile jt has landed
        __syncthreads();        // all copies visible; prior buffer reads done
        if (jt + 1 < CC / 16) {
            async_copy16(kh + (size_t)(jt + 1) * 16 * DD + tid * 4,
                         &ktile[cur ^ 1][tid * 4]);
        }
#else
        __syncthreads();
        ((float4*)ktile[cur])[tid] =
            ((const float4*)(kh + (size_t)jt * 16 * DD))[tid];
        if (jt + 1 < CC / 16)
            __builtin_prefetch(kh + (size_t)(jt + 1) * 16 * DD + tid * 4, 0, 0);
        __syncthreads();
#endif

        v8f c = {};
#pragma unroll
        for (int kk = 0; kk < 16; ++kk) {
            v2f bfrag = *(const v2f*)(&ktile[cur][n * DD + 4 * kk + 2 * half]);
            c = wmma_f32x4(afrag[kk], bfrag, c);
        }

        const int jb = jt * 16 + n;
#pragma unroll
        for (int r = 0; r < 8; ++r) {
            const float val = c[r];              // row M=r+8*half, col jb
            if (val > best[r]) { best[r] = val; bestj[r] = jb; }
        }
    }

    // Cross-lane argmax within each 16-lane half (first-max tie rule).
#pragma unroll
    for (int r = 0; r < 8; ++r) {
        float v = best[r];
        int   j = bestj[r];
#pragma unroll
        for (int m = 8; m >= 1; m >>= 1) {
            const float ov = __shfl_xor(v, m, 32);
            const int   oj = __shfl_xor(j, m, 32);
            if (ov > v || (ov == v && oj < j)) { v = ov; j = oj; }
        }
        bestj[r] = j;
    }

    // Gather winning v rows; write out / indices / histogram.
    const float* vh = vbuf + (size_t)h * CC * DD;
#pragma unroll
    for (int r = 0; r < 8; ++r) {
        const int row = rowbase + r + 8 * half;
        const int j   = bestj[r];                // uniform within half
        const float4 val = ((const float4*)(vh + (size_t)j * DD))[n];
        float4* op = (float4*)(out + (((size_t)b * NN + row) * (HH * DD) +
                                      (size_t)h * DD));
        op[n] = val;
        if (n == 0) {
            idx_out[((size_t)b * HH + h) * NN + row] = (float)j;
            atomicAdd(&hist[h * CC + j], 1u);
        }
    }
}

// ---------------------------------------------------------------------------
// Kernel 3: perplexity per head: exp(-sum(m * log(m + 1e-10))), m = cnt/(B*N)
// ---------------------------------------------------------------------------
__global__ __launch_bounds__(256) void memcodes_perp(
    const unsigned int* __restrict__ hist, float* __restrict__ perp)
{
    __shared__ float red[256];
    const int h   = blockIdx.x;
    const int tid = threadIdx.x;
    float acc = 0.0f;
    const float inv = 1.0f / (float)(BB * NN);
    for (int j = tid; j < CC; j += 256) {
        const float m = (float)hist[h * CC + j] * inv;
        acc += m * __logf(m + 1e-10f);
    }
    red[tid] = acc;
    __syncthreads();
    for (int s = 128; s > 0; s >>= 1) {
        if (tid < s) red[tid] += red[tid + s];
        __syncthreads();
    }
    if (tid == 0) perp[h] = __expf(-red[0]);
}

// ---------------------------------------------------------------------------
extern "C" void kernel_launch(void* const* d_in, const int* in_sizes, int n_in,
                              void* d_out, int out_size, void* d_ws,
                              size_t ws_size, hipStream_t stream) {
    (void)in_sizes; (void)n_in; (void)out_size; (void)ws_size;
    const float* x  = (const float*)d_in[0];   // (B, N, H*D)
    const float* cb = (const float*)d_in[1];   // (H, C, D)
    const float* wk = (const float*)d_in[2];   // (H, D, D)
    const float* wv = (const float*)d_in[3];   // (H, D, D)

    float* out     = (float*)d_out;                          // B*N*H*D
    float* idx_out = out + (size_t)BB * NN * HH * DD;        // B*H*N
    float* perp    = idx_out + (size_t)BB * HH * NN;         // H

    float* kbuf = (float*)d_ws;                              // H*C*D
    float* vbuf = kbuf + (size_t)HH * CC * DD;               // H*C*D
    unsigned int* hist = (unsigned int*)(vbuf + (size_t)HH * CC * DD); // H*C

    (void)hipMemsetAsync(hist, 0, (size_t)HH * CC * sizeof(unsigned int),
                         stream);

    memcodes_project<<<dim3(HH * 64 / 8), 256, 0, stream>>>(cb, wk, wv, kbuf,
                                                            vbuf);
    memcodes_main<<<dim3(NN / 128, HH, BB), 256, 0, stream>>>(x, kbuf, vbuf,
                                                              out, idx_out,
                                                              hist);
    memcodes_perp<<<dim3(HH), 256, 0, stream>>>(hist, perp);
}